// AxisAligndProjectionRenderer_30425548325409
// MI455X (gfx1250) — compile-verified
//
#include <hip/hip_runtime.h>

typedef float v2f __attribute__((ext_vector_type(2)));
typedef float v8f __attribute__((ext_vector_type(8)));

#define RRES 128
#define RR   (RRES*RRES)          /* 16384 samples per batch, coarse */
#define VS   64
#define VT   32
#define PEEP_PER_B (VS*VS*VT)     /* 131072 samples per batch, peep  */
#define NB   4
#define KDIM 96
#define NOUT 35
#define NPAD 48
#define HW   65536                /* 256*256 */
#define SF_STRIDE 98              /* even -> 8B-aligned v2f A-frags; 34s%64 distinct -> no bank conflicts */

// Bilinear setup for grid_sample(align_corners=False, zeros padding), H=W=256.
__device__ __forceinline__ void bilin_setup(float X, float Y,
                                            int& o00, int& o01, int& o10, int& o11,
                                            float& w00, float& w01, float& w10, float& w11)
{
    const float fx = (X + 1.0f) * 128.0f - 0.5f;
    const float fy = (Y + 1.0f) * 128.0f - 0.5f;
    const float x0f = floorf(fx), y0f = floorf(fy);
    const int x0 = (int)x0f, y0 = (int)y0f;
    const int x1 = x0 + 1,  y1 = y0 + 1;
    const float wx1 = fx - x0f, wx0 = 1.0f - wx1;
    const float wy1 = fy - y0f, wy0 = 1.0f - wy1;
    const float mx0 = (x0 >= 0 && x0 < 256) ? 1.0f : 0.0f;
    const float mx1 = (x1 >= 0 && x1 < 256) ? 1.0f : 0.0f;
    const float my0 = (y0 >= 0 && y0 < 256) ? 1.0f : 0.0f;
    const float my1 = (y1 >= 0 && y1 < 256) ? 1.0f : 0.0f;
    const int x0c = min(max(x0, 0), 255), x1c = min(max(x1, 0), 255);
    const int y0c = min(max(y0, 0), 255), y1c = min(max(y1, 0), 255);
    w00 = wy0 * wx0 * my0 * mx0;  w01 = wy0 * wx1 * my0 * mx1;
    w10 = wy1 * wx0 * my1 * mx0;  w11 = wy1 * wx1 * my1 * mx1;
    o00 = (y0c << 8) + x0c;  o01 = (y0c << 8) + x1c;
    o10 = (y1c << 8) + x0c;  o11 = (y1c << 8) + x1c;
}

// Sample coordinates for sample `loc` of batch `b`. Returns (c0,c1,c2) and which
// plane is constant across a 16-aligned tile (loc's fastest-varying index only).
__device__ __forceinline__ void coords_of(int mode, int b, int loc, const float* __restrict__ cvec,
                                          float& c0, float& c1, float& c2, int& constPlane)
{
    if (mode == 0) {
        const int i = loc >> 7, j = loc & 127;
        const float gx = -1.0f + (2.0f / 127.0f) * (float)i;
        const float gy = -1.0f + (2.0f / 127.0f) * (float)j;     // varies within tile
        const float t  = cvec[b * 6 + 3] * 2.0f - 1.0f;
        const float a0 = cvec[b * 6 + 0], a1 = cvec[b * 6 + 1], a2 = cvec[b * 6 + 2];
        int idx = 0; float best = a0;
        if (a1 > best) { idx = 1; best = a1; }
        if (a2 > best) { idx = 2; }
        if (idx == 0)      { c0 = t;  c1 = gx; c2 = gy; }
        else if (idx == 1) { c0 = gx; c1 = t;  c2 = gy; }
        else               { c0 = gx; c1 = gy; c2 = t;  }
        constPlane = (idx == 2) ? 1 : 0;   // the plane not touching the varying coord
    } else {
        const int i = loc >> 11;           // varies per tile only
        const int j = (loc >> 5) & 63;
        const int t = loc & 31;            // varies within tile
        c0 = (cvec[b * 6 + 4] * 2.0f - 1.0f) + (2.0f / 63.0f) * (float)i;
        c1 = (cvec[b * 6 + 5] * 2.0f - 1.0f) + (2.0f / 63.0f) * (float)j;
        c2 = -1.0f + (2.0f / 31.0f) * (float)t;
        constPlane = 0;                    // plane0 = (c0,c1) independent of t
    }
}

// mode 0: coarse pass (B*R*R samples); mode 1: peep-video pass (B*Vs*Vs*Vt samples)
__global__ __launch_bounds__(128)
void triplane_render_kernel(const float* __restrict__ planes,
                            const float* __restrict__ cvec,
                            const float* __restrict__ Wd,
                            const float* __restrict__ bd,
                            float* __restrict__ outColors,
                            float* __restrict__ outPeep,
                            float* __restrict__ outFeat,
                            int mode)
{
    // W_dec K-pair interleaved: sWdB[k/2][n] = {Wd[k][n], Wd[k+1][n]} -> one ds_load_b64 / B-frag
    __shared__ float sWdB[KDIM / 2][NPAD][2];
    __shared__ float sSf[4][16][SF_STRIDE];   // per-wave 16-sample x 96-feature tile

    const int tid  = threadIdx.x;
    const int lane = tid & 31;
    const int wave = tid >> 5;

    // ---- stage decoder weights, zero-padded to 48 cols, K-pair interleaved ----
    for (int i = tid; i < KDIM * NPAD; i += 128) {
        const int k = i / NPAD, n = i - k * NPAD;
        sWdB[k >> 1][n][k & 1] = (n < NOUT) ? Wd[k * NOUT + n] : 0.0f;
    }

    const int perB     = mode ? PEEP_PER_B : RR;
    const int tileBase = blockIdx.x * 64 + wave * 16;   // first sample of this wave's tile
    const int b        = tileBase / perB;               // uniform per wave (perB % 64 == 0)
    const int loc0     = tileBase - b * perB;
    const float* planesB = planes + (size_t)b * (size_t)(3 * 32 * HW);

    // ---- phase A: the tile-constant plane, computed once per wave (lane = channel) ----
    int cp;
    {
        float c0, c1, c2;
        coords_of(mode, b, loc0, cvec, c0, c1, c2, cp);
        const float PX[3] = { c0, c0, c2 };
        const float PY[3] = { c1, c2, c1 };
        int o00, o01, o10, o11; float w00, w01, w10, w11;
        bilin_setup(PX[cp], PY[cp], o00, o01, o10, o11, w00, w01, w10, w11);
        const float* f = planesB + (size_t)cp * (32 * HW) + (size_t)lane * HW;  // lane = channel
        const float val = f[o00] * w00 + f[o01] * w01 + f[o10] * w10 + f[o11] * w11;
        #pragma unroll
        for (int s2 = 0; s2 < 16; ++s2)
            sSf[wave][s2][cp * 32 + lane] = val;     // broadcast to all 16 samples
    }

    // ---- phase B: the two varying planes, lanes = (sample s, channel half) ----
    {
        const int s    = lane & 15;
        const int half = lane >> 4;
        const int chBeg = half * 16, chEnd = chBeg + 16;
        float c0, c1, c2; int dummy;
        coords_of(mode, b, loc0 + s, cvec, c0, c1, c2, dummy);
        const float PX[3] = { c0, c0, c2 };
        const float PY[3] = { c1, c2, c1 };
        for (int p = 0; p < 3; ++p) {
            if (p == cp) continue;                   // uniform branch per wave
            int o00, o01, o10, o11; float w00, w01, w10, w11;
            bilin_setup(PX[p], PY[p], o00, o01, o10, o11, w00, w01, w10, w11);
            const float* fb = planesB + (size_t)p * (32 * HW);
            float* dst = &sSf[wave][s][p * 32];
            #pragma unroll 4
            for (int ch = chBeg; ch < chEnd; ++ch) {
                const float* f = fb + (size_t)ch * HW;
                dst[ch] = f[o00] * w00 + f[o01] * w01 + f[o10] * w10 + f[o11] * w11;
            }
        }
    }
    __syncthreads();

    // ---- decode: D[16x35] = sf[16x96] @ W_dec[96x35] + b_dec via V_WMMA_F32_16X16X4_F32 ----
    const int row = lane & 15;        // A-matrix M index
    const int kp0 = lane >> 4;        // lanes 16-31 hold the K+2,K+3 sub-columns
    const int col = lane & 15;        // B/D N index within tile

    v8f acc[3];
    #pragma unroll
    for (int nt = 0; nt < 3; ++nt) {
        const int n = nt * 16 + col;
        const float bv = (n < NOUT) ? bd[n] : 0.0f;
        v8f t = { bv, bv, bv, bv, bv, bv, bv, bv };
        acc[nt] = t;
    }

    #pragma unroll 4
    for (int k4 = 0; k4 < KDIM / 4; ++k4) {
        const int kb = 4 * k4 + 2 * kp0;                       // even -> 8B aligned
        const v2f a = *(const v2f*)&sSf[wave][row][kb];        // ds_load_b64
        #pragma unroll
        for (int nt = 0; nt < 3; ++nt) {
            const v2f bm = *(const v2f*)&sWdB[2 * k4 + kp0][nt * 16 + col][0];  // ds_load_b64
            acc[nt] = __builtin_amdgcn_wmma_f32_16x16x4_f32(
                false, a, false, bm, (short)0, acc[nt], false, false);
        }
    }

    // ---- scatter-store D fragment (lane holds N=col, M=r + 8*(lane>=16)) ----
    const int mrow = kp0 * 8;
    #pragma unroll
    for (int nt = 0; nt < 3; ++nt) {
        const int n = nt * 16 + col;
        if (n >= NOUT) continue;
        #pragma unroll
        for (int r = 0; r < 8; ++r) {
            const float v = acc[nt][r];
            const int g = tileBase + mrow + r;     // global sample index in this pass
            if (mode == 0) {
                if (n < 3) outColors[(size_t)g * 3 + n] = v;            // [B*RR, 3]
                else       outFeat[(size_t)g * 32 + (n - 3)] = v;       // [B*RR, 32]
            } else {
                if (n < 3) {
                    const int bb  = g >> 17;                 // / 131072
                    const int loc = g & (PEEP_PER_B - 1);
                    outPeep[(size_t)(bb * 3 + n) * PEEP_PER_B + loc] = v; // [B,3,Vs,Vs,Vt]
                }
            }
        }
    }
}

extern "C" void kernel_launch(void* const* d_in, const int* in_sizes, int n_in,
                              void* d_out, int out_size, void* d_ws, size_t ws_size,
                              hipStream_t stream) {
    const float* planes = (const float*)d_in[0];   // [4,3,32,256,256] f32
    const float* cvec   = (const float*)d_in[1];   // [4,6] f32
    const float* Wd     = (const float*)d_in[2];   // [96,35] f32
    const float* bd     = (const float*)d_in[3];   // [35] f32
    // d_in[4] plane_axes: fixed permutations, folded into the kernel
    // d_in[5] resolution: 128, folded into the kernel

    float* out       = (float*)d_out;
    float* outColors = out;                                        // 4*16384*3
    float* outPeep   = out + (size_t)NB * RR * 3;                  // 4*3*131072
    float* outFeat   = outPeep + (size_t)NB * 3 * PEEP_PER_B;      // 4*16384*32

    // coarse pass: 65536 samples, 64 per block (1024 blocks)
    triplane_render_kernel<<<(NB * RR) / 64, 128, 0, stream>>>(
        planes, cvec, Wd, bd, outColors, outPeep, outFeat, 0);
    // peep-video pass: 524288 samples (8192 blocks)
    triplane_render_kernel<<<(NB * PEEP_PER_B) / 64, 128, 0, stream>>>(
        planes, cvec, Wd, bd, outColors, outPeep, outFeat, 1);
}